// MedianFilter_22737556865485
// MI455X (gfx1250) — compile-verified
//
#include <hip/hip_runtime.h>
#include <hip/hip_bf16.h>

// 3x3 median blur (kornia median_blur semantics: reflect-101 pad, sorted[4]).
// Input/Output: [16, 3, 512, 512] float32.
//
// Strategy (MI455X / gfx1250, wave32):
//  - Memory-bound op: ~100 MB total traffic -> ~4.3 us floor @ 23.3 TB/s.
//  - Stage (TY+2)x(TX+2) tile into LDS via CDNA5 async global->LDS
//    (GLOBAL_LOAD_ASYNC_TO_LDS_B32, drained by s_wait_asynccnt 0) to avoid
//    the 9x global read amplification.
//  - median9 via column triples + v_min3/v_med3/v_max3:
//       per column c: lo=min3, mi=med3, hi=max3 of the 3 rows
//       median = med3( max3(lo,lo,lo), med3(mi,mi,mi), min3(hi,hi,hi) )
//    Column triples shared across 4 horizontally adjacent outputs per thread
//    -> 34 three-input VALU ops per 4 pixels.
//  - Each lane writes an aligned float4 (coalesced global_store_b128).

#define IMG_W 512
#define IMG_H 512
#define TX 128          // tile width  (outputs per block, x)
#define TY 8            // tile height (outputs per block, y)
#define LW (TX + 2)     // staged width  (with halo)
#define LH (TY + 2)     // staged height (with halo)
#define NTHREADS 256    // 8 waves of 32

#if defined(__gfx1250__) && __has_builtin(__builtin_amdgcn_global_load_async_to_lds_b32) && __has_builtin(__builtin_amdgcn_s_wait_asynccnt)
#define USE_ASYNC_LDS 1
typedef __attribute__((address_space(1))) int* gptr_t;
typedef __attribute__((address_space(3))) int* lptr_t;
#else
#define USE_ASYNC_LDS 0
#endif

__device__ __forceinline__ float min3f(float a, float b, float c) {
    return fminf(fminf(a, b), c);
}
__device__ __forceinline__ float max3f(float a, float b, float c) {
    return fmaxf(fmaxf(a, b), c);
}
__device__ __forceinline__ float med3f(float a, float b, float c) {
#if defined(__AMDGCN__)
    return __builtin_amdgcn_fmed3f(a, b, c);
#else
    return fmaxf(fminf(fmaxf(a, b), c), fminf(a, b));
#endif
}

__global__ __launch_bounds__(NTHREADS)
void median3x3_kernel(const float* __restrict__ in, float* __restrict__ out) {
    __shared__ float tile[LH][LW];

    const int x0 = blockIdx.x * TX;
    const int y0 = blockIdx.y * TY;
    const long plane = (long)blockIdx.z * (long)(IMG_H * IMG_W);
    const float* __restrict__ src = in + plane;
    const int tid = threadIdx.x;

    // ---- Stage input tile + halo into LDS (reflect-101 at image borders) ----
    for (int i = tid; i < LH * LW; i += NTHREADS) {
        const int ly = i / LW;
        const int lx = i - ly * LW;
        int gy = y0 - 1 + ly;
        int gx = x0 - 1 + lx;
        gy = (gy < 0) ? 1 : ((gy >= IMG_H) ? (IMG_H - 2) : gy);
        gx = (gx < 0) ? 1 : ((gx >= IMG_W) ? (IMG_W - 2) : gx);
#if USE_ASYNC_LDS
        __builtin_amdgcn_global_load_async_to_lds_b32(
            (gptr_t)(src + gy * IMG_W + gx),
            (lptr_t)&tile[ly][lx],
            /*offset=*/0, /*cpol=*/0);
#else
        tile[ly][lx] = src[gy * IMG_W + gx];
#endif
    }
#if USE_ASYNC_LDS
    __builtin_amdgcn_s_wait_asynccnt(0);
#endif
    __syncthreads();

    // ---- Compute: one wave per tile row, 4 outputs per lane ----
    const int lane = tid & 31;   // 0..31
    const int trow = tid >> 5;   // 0..7 : output row within tile
    const int cx = lane << 2;    // 0,4,...,124 : leftmost halo column of this lane

    float lo[6], mi[6], hi[6];
#pragma unroll
    for (int c = 0; c < 6; ++c) {
        const float a = tile[trow][cx + c];
        const float b = tile[trow + 1][cx + c];
        const float d = tile[trow + 2][cx + c];
        lo[c] = min3f(a, b, d);
        mi[c] = med3f(a, b, d);
        hi[c] = max3f(a, b, d);
    }

    float4 r;
    float* rp = &r.x;
#pragma unroll
    for (int j = 0; j < 4; ++j) {
        const float L = max3f(lo[j], lo[j + 1], lo[j + 2]);
        const float M = med3f(mi[j], mi[j + 1], mi[j + 2]);
        const float U = min3f(hi[j], hi[j + 1], hi[j + 2]);
        rp[j] = med3f(L, M, U);
    }

    float* __restrict__ dst = out + plane + (long)(y0 + trow) * IMG_W + (x0 + cx);
    *reinterpret_cast<float4*>(dst) = r;   // 16B-aligned: x0+cx is a multiple of 4
}

extern "C" void kernel_launch(void* const* d_in, const int* in_sizes, int n_in,
                              void* d_out, int out_size, void* d_ws, size_t ws_size,
                              hipStream_t stream) {
    (void)in_sizes; (void)n_in; (void)out_size; (void)d_ws; (void)ws_size;
    const float* in = (const float*)d_in[0];
    float* out = (float*)d_out;
    // 16 batches * 3 channels = 48 planes of 512x512; tiles of 128x8.
    dim3 grid(IMG_W / TX, IMG_H / TY, 16 * 3);
    median3x3_kernel<<<grid, NTHREADS, 0, stream>>>(in, out);
}